// CayleyConv_54193897341473
// MI455X (gfx1250) — compile-verified
//
#include <hip/hip_runtime.h>

// ---------------------------------------------------------------------------
// CayleyConv on MI455X (gfx1250, wave32):
//   * Sparse edge-scatter Jacobi solver (global_atomic_add_f32, L2-resident)
//   * fp32 WMMA (v_wmma_f32_16x16x4_f32) for the 4096x64x64 weight GEMMs
// ---------------------------------------------------------------------------

#define FDIM 64

typedef float v2f __attribute__((ext_vector_type(2)));
typedef float v8f __attribute__((ext_vector_type(8)));

// ---------------------------- elementwise / setup --------------------------

__global__ void k_init(float* __restrict__ deg, float* __restrict__ diagL,
                       float* __restrict__ cum, int N, int NF) {
  int i = blockIdx.x * blockDim.x + threadIdx.x;
  if (i < N) { deg[i] = 0.0f; diagL[i] = 1.0f; }
  if (i < NF) cum[i] = 0.0f;
}

__global__ void k_deg(const long long* __restrict__ ei, const float* __restrict__ w,
                      float* __restrict__ deg, int E) {
  int e = blockIdx.x * blockDim.x + threadIdx.x;
  if (e >= E) return;
  int r = (int)ei[e];
  unsafeAtomicAdd(&deg[r], w[e]);
}

__global__ void k_dinv(const float* __restrict__ deg, float* __restrict__ dinv, int N) {
  int n = blockIdx.x * blockDim.x + threadIdx.x;
  if (n >= N) return;
  float d = deg[n];
  dinv[n] = (d > 0.0f) ? (1.0f / sqrtf(d)) : 0.0f;
}

__global__ void k_selfloop(const long long* __restrict__ ei, const float* __restrict__ w,
                           const float* __restrict__ dinv, float* __restrict__ diagL, int E) {
  int e = blockIdx.x * blockDim.x + threadIdx.x;
  if (e >= E) return;
  int r = (int)ei[e];
  int c = (int)ei[E + e];
  if (r == c) unsafeAtomicAdd(&diagL[r], -dinv[r] * dinv[c] * w[e]);
}

// complex diag of A = h*diagL + i;  1/(a+i) = (a - i)/(a^2+1)
__global__ void k_dinvD(const float* __restrict__ diagL, const float* __restrict__ h,
                        float* __restrict__ ddr, float* __restrict__ ddi, int N) {
  int n = blockIdx.x * blockDim.x + threadIdx.x;
  if (n >= N) return;
  float a = h[0] * diagL[n];
  float inv = 1.0f / (a * a + 1.0f);
  ddr[n] = a * inv;
  ddi[n] = -inv;
}

__global__ void k_edgecoef(const long long* __restrict__ ei, const float* __restrict__ w,
                           const float* __restrict__ dinv, int* __restrict__ rows,
                           int* __restrict__ cols, float* __restrict__ cAll,
                           float* __restrict__ cOff, int E) {
  int e = blockIdx.x * blockDim.x + threadIdx.x;
  if (e >= E) return;
  int r = (int)ei[e];
  int c = (int)ei[E + e];
  float s = dinv[r] * dinv[c] * w[e];
  rows[e] = r; cols[e] = c;
  cAll[e] = s;
  cOff[e] = (r == c) ? 0.0f : s;   // off-diagonal operator: self-loops cancel exactly
}

__global__ void k_inity(const float* __restrict__ x, float* __restrict__ Yre,
                        float* __restrict__ Yim, int NF) {
  int i = blockIdx.x * blockDim.x + threadIdx.x;
  if (i >= NF) return;
  Yre[i] = x[i];
  Yim[i] = 0.0f;
}

__global__ void k_zeroS(float* __restrict__ Sre, float* __restrict__ Sim, int NF) {
  int i = blockIdx.x * blockDim.x + threadIdx.x;
  if (i >= NF) return;
  Sre[i] = 0.0f;
  Sim[i] = 0.0f;
}

// S[r,:] += s_e * X[c,:]   (both complex planes), one thread per (edge,feature)
__global__ void k_scatter(const int* __restrict__ rows, const int* __restrict__ cols,
                          const float* __restrict__ coef,
                          const float* __restrict__ Xre, const float* __restrict__ Xim,
                          float* __restrict__ Sre, float* __restrict__ Sim, int E) {
  int t = blockIdx.x * blockDim.x + threadIdx.x;
  if (t >= E * FDIM) return;
  int e = t >> 6;
  int f = t & (FDIM - 1);
  int r = rows[e], c = cols[e];
  float s = coef[e];
  unsafeAtomicAdd(&Sre[r * FDIM + f], s * Xre[c * FDIM + f]);
  unsafeAtomicAdd(&Sim[r * FDIM + f], s * Xim[c * FDIM + f]);
}

// b = (hL - iI) y = h*(y - Sy) - i*y ;  d = dinvD*b ; x0 = b (written in-place into Y)
__global__ void k_bdx(float* __restrict__ Yre, float* __restrict__ Yim,
                      const float* __restrict__ Sre, const float* __restrict__ Sim,
                      const float* __restrict__ ddr, const float* __restrict__ ddi,
                      const float* __restrict__ h,
                      float* __restrict__ Dre, float* __restrict__ Dim, int NF) {
  int i = blockIdx.x * blockDim.x + threadIdx.x;
  if (i >= NF) return;
  int n = i >> 6;
  float hv = h[0];
  float yr = Yre[i], yi = Yim[i];
  float br = hv * (yr - Sre[i]) + yi;
  float bi = hv * (yi - Sim[i]) - yr;
  float dr = ddr[n], di = ddi[n];
  Dre[i] = dr * br - di * bi;
  Dim[i] = dr * bi + di * br;
  Yre[i] = br;
  Yim[i] = bi;
}

// x_{k+1} = d + dinvD * (h * S')   (S' built from off-diagonal coefficients)
__global__ void k_update(const float* __restrict__ Sre, const float* __restrict__ Sim,
                         const float* __restrict__ Dre, const float* __restrict__ Dim,
                         const float* __restrict__ ddr, const float* __restrict__ ddi,
                         const float* __restrict__ h,
                         float* __restrict__ Xre, float* __restrict__ Xim, int NF) {
  int i = blockIdx.x * blockDim.x + threadIdx.x;
  if (i >= NF) return;
  int n = i >> 6;
  float hv = h[0];
  float tre = hv * Sre[i];
  float tim = hv * Sim[i];
  float dr = ddr[n], di = ddi[n];
  Xre[i] = Dre[i] + dr * tre - di * tim;
  Xim[i] = Dim[i] + dr * tim + di * tre;
}

// ------------------------------- WMMA GEMMs --------------------------------
// One wave computes one 16x16 output tile; K=64 consumed 4 at a time with
// v_wmma_f32_16x16x4_f32. Wave32 layouts per CDNA5 ISA 7.12.2:
//   A (16x4):  lanes 0-15 -> M=lane, K={k0,k0+1}; lanes 16-31 -> K={k0+2,k0+3}
//   B (4x16):  lanes 0-15 -> N=lane, K={k0,k0+1}; lanes 16-31 -> K={k0+2,k0+3}
//   C (16x16): vgpr j -> M = j + 8*half, N = lane&15

__device__ __forceinline__ v8f wmma4(v2f a, v2f b, v8f c) {
  return __builtin_amdgcn_wmma_f32_16x16x4_f32(false, a, false, b, (short)0, c,
                                               false, false);
}

// Cum += Yre @ Wre - Yim @ Wim      (real part of complex y@Wc)
__global__ void k_gemm_cum(const float* __restrict__ Yre, const float* __restrict__ Yim,
                           const float* __restrict__ Wre, const float* __restrict__ Wim,
                           float* __restrict__ Cum) {
  int wave = threadIdx.x >> 5;
  int lane = threadIdx.x & 31;
  int tile = blockIdx.x * 4 + wave;
  int tileM = tile >> 2;          // FDIM/16 = 4 tiles along output columns
  int tileN = tile & 3;
  int half = lane >> 4;
  int l = lane & 15;
  int rowA = tileM * 16 + l;
  int colB = tileN * 16 + l;

  v8f c;
#pragma unroll
  for (int j = 0; j < 8; ++j)
    c[j] = Cum[(tileM * 16 + j + 8 * half) * FDIM + tileN * 16 + l];

#pragma unroll
  for (int k0 = 0; k0 < FDIM; k0 += 4) {
    int ka = k0 + 2 * half;
    v2f a; a[0] = Yre[rowA * FDIM + ka]; a[1] = Yre[rowA * FDIM + ka + 1];
    v2f b; b[0] = Wre[ka * FDIM + colB]; b[1] = Wre[(ka + 1) * FDIM + colB];
    c = wmma4(a, b, c);
  }
#pragma unroll
  for (int k0 = 0; k0 < FDIM; k0 += 4) {   // minus Yim@Wim: negate B on load
    int ka = k0 + 2 * half;
    v2f a; a[0] = Yim[rowA * FDIM + ka]; a[1] = Yim[rowA * FDIM + ka + 1];
    v2f b; b[0] = -Wim[ka * FDIM + colB]; b[1] = -Wim[(ka + 1) * FDIM + colB];
    c = wmma4(a, b, c);
  }
#pragma unroll
  for (int j = 0; j < 8; ++j)
    Cum[(tileM * 16 + j + 8 * half) * FDIM + tileN * 16 + l] = c[j];
}

// out = x @ W0 + 2*Cum
__global__ void k_gemm_out(const float* __restrict__ X, const float* __restrict__ W0,
                           const float* __restrict__ Cum, float* __restrict__ Out) {
  int wave = threadIdx.x >> 5;
  int lane = threadIdx.x & 31;
  int tile = blockIdx.x * 4 + wave;
  int tileM = tile >> 2;
  int tileN = tile & 3;
  int half = lane >> 4;
  int l = lane & 15;
  int rowA = tileM * 16 + l;
  int colB = tileN * 16 + l;

  v8f c = {};
#pragma unroll
  for (int k0 = 0; k0 < FDIM; k0 += 4) {
    int ka = k0 + 2 * half;
    v2f a; a[0] = X[rowA * FDIM + ka]; a[1] = X[rowA * FDIM + ka + 1];
    v2f b; b[0] = W0[ka * FDIM + colB]; b[1] = W0[(ka + 1) * FDIM + colB];
    c = wmma4(a, b, c);
  }
#pragma unroll
  for (int j = 0; j < 8; ++j) {
    int idx = (tileM * 16 + j + 8 * half) * FDIM + tileN * 16 + l;
    Out[idx] = c[j] + 2.0f * Cum[idx];
  }
}

// ------------------------------- launch ------------------------------------

extern "C" void kernel_launch(void* const* d_in, const int* in_sizes, int n_in,
                              void* d_out, int out_size, void* d_ws, size_t ws_size,
                              hipStream_t stream) {
  const float*     x    = (const float*)d_in[0];
  const long long* ei   = (const long long*)d_in[1];   // int64 (2,E)
  const float*     w    = (const float*)d_in[2];
  const float*     h    = (const float*)d_in[3];
  const float*     W0   = (const float*)d_in[4];
  const float*     Wcre = (const float*)d_in[5];       // (3,64,64)
  const float*     Wcim = (const float*)d_in[6];

  const int E  = in_sizes[2];
  const int N  = in_sizes[0] / FDIM;
  const int NF = N * FDIM;

  char* cur = (char*)d_ws;
  auto carve = [&](size_t bytes) {
    void* p = (void*)cur;
    cur += (bytes + 255) & ~(size_t)255;
    return p;
  };
  float* deg   = (float*)carve((size_t)N * 4);
  float* dinv  = (float*)carve((size_t)N * 4);
  float* diagL = (float*)carve((size_t)N * 4);
  float* ddr   = (float*)carve((size_t)N * 4);
  float* ddi   = (float*)carve((size_t)N * 4);
  int*   rows  = (int*)  carve((size_t)E * 4);
  int*   cols  = (int*)  carve((size_t)E * 4);
  float* cAll  = (float*)carve((size_t)E * 4);
  float* cOff  = (float*)carve((size_t)E * 4);
  float* Yre   = (float*)carve((size_t)NF * 4);
  float* Yim   = (float*)carve((size_t)NF * 4);
  float* Sre   = (float*)carve((size_t)NF * 4);
  float* Sim   = (float*)carve((size_t)NF * 4);
  float* Dre   = (float*)carve((size_t)NF * 4);
  float* Dim   = (float*)carve((size_t)NF * 4);
  float* Cum   = (float*)carve((size_t)NF * 4);

  const int gN  = (N + 255) / 256;
  const int gE  = (E + 255) / 256;
  const int gNF = (NF + 255) / 256;
  const int gEF = (E * FDIM + 255) / 256;
  const int gGemm = (N / 16) * (FDIM / 16) / 4;   // 4 waves (16x16 tiles) per block

  k_init    <<<gNF, 256, 0, stream>>>(deg, diagL, Cum, N, NF);
  k_deg     <<<gE,  256, 0, stream>>>(ei, w, deg, E);
  k_dinv    <<<gN,  256, 0, stream>>>(deg, dinv, N);
  k_selfloop<<<gE,  256, 0, stream>>>(ei, w, dinv, diagL, E);
  k_dinvD   <<<gN,  256, 0, stream>>>(diagL, h, ddr, ddi, N);
  k_edgecoef<<<gE,  256, 0, stream>>>(ei, w, dinv, rows, cols, cAll, cOff, E);
  k_inity   <<<gNF, 256, 0, stream>>>(x, Yre, Yim, NF);

  for (int term = 0; term < 3; ++term) {
    // b = (hL - iI) y ; d = dinvD*b ; x0 = b
    k_zeroS  <<<gNF, 256, 0, stream>>>(Sre, Sim, NF);
    k_scatter<<<gEF, 256, 0, stream>>>(rows, cols, cAll, Yre, Yim, Sre, Sim, E);
    k_bdx    <<<gNF, 256, 0, stream>>>(Yre, Yim, Sre, Sim, ddr, ddi, h, Dre, Dim, NF);
    // 10 Jacobi sweeps: x <- d + dinvD*(h*S_off(x))
    for (int it = 0; it < 10; ++it) {
      k_zeroS  <<<gNF, 256, 0, stream>>>(Sre, Sim, NF);
      k_scatter<<<gEF, 256, 0, stream>>>(rows, cols, cOff, Yre, Yim, Sre, Sim, E);
      k_update <<<gNF, 256, 0, stream>>>(Sre, Sim, Dre, Dim, ddr, ddi, h, Yre, Yim, NF);
    }
    // Cum += Re(y @ Wc[term])
    k_gemm_cum<<<gGemm, 128, 0, stream>>>(Yre, Yim,
                                          Wcre + (size_t)term * FDIM * FDIM,
                                          Wcim + (size_t)term * FDIM * FDIM, Cum);
  }
  k_gemm_out<<<gGemm, 128, 0, stream>>>(x, W0, Cum, (float*)d_out);
}